// Capsule_2224793059594
// MI455X (gfx1250) — compile-verified
//
#include <hip/hip_runtime.h>
#include <math.h>

// ---------------------------------------------------------------------------
// Capsule routing, factored form.
//   u_hat[b,s,n,d] = UW[b,s,d] + usum[b,s]*pe1[n,d] + pe2[s, n*64+d]
// Routing contractions distribute over this sum -> never materialize u_hat.
// ---------------------------------------------------------------------------

#define NCAP   128
#define DCAP   64
#define STEPS  128
#define INDIM  256
#define BATCH  32
#define EPSF   1e-7f

typedef __attribute__((ext_vector_type(2))) float v2f;
typedef __attribute__((ext_vector_type(8))) float v8f;

// D = A(16x4,f32) * B(4x16,f32) + C(16x16,f32)  -- wave32, CDNA5
static __device__ inline v8f wmma_f32(v2f a, v2f b, v8f c) {
    return __builtin_amdgcn_wmma_f32_16x16x4_f32(
        false, a, false, b, (short)0, c, false, false);
}

// sinusoidal table value for a (S, D) embedding
static __device__ inline float pe_val(int s, int d, float Df) {
    int k = d >> 1;
    // 1/10000^(2k/D) = exp(-(2k/D)*ln(10000))
    float inv = __expf(-(2.0f * (float)k / Df) * 9.210340371976184f);
    float ang = (float)s * inv;
    return (d & 1) ? __cosf(ang) : __sinf(ang);
}

// pe1: (NCAP=128, DCAP=64) table (the kernel-side pos_emb; its S index is n)
__global__ void __launch_bounds__(256) pe1_kernel(float* __restrict__ pe1) {
    int i = blockIdx.x * 256 + threadIdx.x;
    if (i < NCAP * DCAP) {
        int s = i >> 6, d = i & 63;
        pe1[i] = pe_val(s, d, (float)DCAP);
    }
}

// pe2: (STEPS=128, N*D=8192) table (the u_hat-side pos_emb)
__global__ void __launch_bounds__(256) pe2_kernel(float* __restrict__ pe2) {
    int i = blockIdx.x * 256 + threadIdx.x;
    if (i < STEPS * NCAP * DCAP) {
        int s = i >> 13, o = i & 8191;
        pe2[i] = pe_val(s, o, (float)(NCAP * DCAP));
    }
}

// UW = U(4096x256) @ W(256x64), plus usum = row sums of U.
// One wave per 16-row stripe; 4 WMMA accumulators cover the 64 output cols.
__global__ void __launch_bounds__(32) gemm_uw_kernel(
    const float* __restrict__ U, const float* __restrict__ W,
    float* __restrict__ UWg, float* __restrict__ usumg) {
    const int row0 = blockIdx.x * 16;
    const int lane = threadIdx.x;
    const int m    = lane & 15;        // M within tile / N within B tile
    const int khi  = lane >> 4;        // 0: K={0,1}, 1: K={2,3}

    v8f acc[4] = {};
    float psum = 0.0f;

    for (int k0 = 0; k0 < INDIM; k0 += 4) {
        const int ka = k0 + 2 * khi;
        v2f a;
        a.x = U[(size_t)(row0 + m) * INDIM + ka];
        a.y = U[(size_t)(row0 + m) * INDIM + ka + 1];
        psum += a.x + a.y;
#pragma unroll
        for (int t = 0; t < 4; ++t) {
            v2f bb;
            bb.x = W[(size_t)ka * DCAP + t * 16 + m];
            bb.y = W[(size_t)(ka + 1) * DCAP + t * 16 + m];
            acc[t] = wmma_f32(a, bb, acc[t]);
        }
    }

    float rs = psum + __shfl_xor(psum, 16, 32);
    if (khi == 0) usumg[row0 + m] = rs;

#pragma unroll
    for (int t = 0; t < 4; ++t)
#pragma unroll
        for (int j = 0; j < 8; ++j)
            UWg[(size_t)(row0 + j + 8 * khi) * DCAP + t * 16 + m] = acc[t][j];
}

// One workgroup (256 threads = 8 waves) per batch element.
// LDS (padded strides to kill bank conflicts):
//   uw   [128][65]   blog [128][129]   pre  [128][65]   outm [128][65]
//   usum [128] msk [128] t2 [128] g2 [128]
__global__ void __launch_bounds__(256) routing_kernel(
    const float* __restrict__ UWg,  const float* __restrict__ usumg,
    const float* __restrict__ pe1g, const float* __restrict__ pe2g,
    const float* __restrict__ maskg, float* __restrict__ out) {
    extern __shared__ float sm[];
    float* uw   = sm;                      // 128*65
    float* blog = uw + 128 * 65;           // 128*129 (logits, then c in-place)
    float* pre  = blog + 128 * 129;        // 128*65
    float* outm = pre + 128 * 65;          // 128*65
    float* usum = outm + 128 * 65;         // 128
    float* msk  = usum + 128;              // 128
    float* t2   = msk + 128;               // 128
    float* g2   = t2 + 128;                // 128

    const int b    = blockIdx.x;
    const int tid  = threadIdx.x;
    const int lane = tid & 31;
    const int wave = tid >> 5;             // 0..7
    const int mrow = lane & 15;
    const int khi  = lane >> 4;

    // ---- Stage UW[b] (128x64 f32) into LDS with stride-65 padding ---------
    // The TDM does the padded tile move in hardware: tile 64x128 dwords,
    // pad_interval=5 (insert pad every 64 DWORDs), pad_amount=0 (1 DWORD).
#if __has_builtin(__builtin_amdgcn_tensor_load_to_lds)
    if (wave == 0) {
        typedef __attribute__((ext_vector_type(4))) unsigned tdm_g0_t;
        typedef __attribute__((ext_vector_type(8))) int      tdm_g1_t;
        typedef __attribute__((ext_vector_type(4))) int      tdm_g2_t;
        typedef __attribute__((ext_vector_type(8))) int      tdm_g4_t;
        unsigned long long ga =
            (unsigned long long)(size_t)(UWg + (size_t)b * STEPS * DCAP);
        unsigned lds_off = (unsigned)(size_t)(void*)uw;  // aperture low 32b
        tdm_g0_t g0;
        g0.x = 1u;                                   // count=1, user D#
        g0.y = lds_off;                              // lds_addr (bytes)
        g0.z = (unsigned)(ga & 0xffffffffu);         // global_addr[31:0]
        g0.w = (unsigned)((ga >> 32) & 0x01ffffffu)  // global_addr[56:32]
             | 0x80000000u;                          // type=2 (tensor)
        tdm_g1_t g1;
        g1[0] = (2 << 16)       // data_size = 4 bytes
              | (1 << 20)       // pad_enable
              | (5 << 22);      // pad_interval: every 64 DWORDs
                                // pad_amount=0 -> 1 DWORD  => LDS stride 65
        g1[1] = (DCAP  << 16);  // tensor_dim0 = 64   (bits 79:48 span)
        g1[2] = (STEPS << 16);  // tensor_dim1 = 128  (bits 111:80 span)
        g1[3] = (DCAP  << 16);  // tile_dim0   = 64   (bits 127:112)
        g1[4] = STEPS;          // tile_dim1   = 128  (bits 143:128)
        g1[5] = DCAP;           // tensor_dim0_stride = 64 (low 32 of 48)
        g1[6] = 0;
        g1[7] = 0;
        tdm_g2_t gz = {0, 0, 0, 0};
        tdm_g4_t gz8 = {0, 0, 0, 0, 0, 0, 0, 0};
        __builtin_amdgcn_tensor_load_to_lds(g0, g1, gz, gz, gz8, 0);
        __builtin_amdgcn_s_wait_tensorcnt(0);
    }
#else
    for (int i = tid; i < STEPS * DCAP; i += 256) {
        int s = i >> 6, d = i & 63;
        uw[s * 65 + d] = UWg[((size_t)b * STEPS + s) * DCAP + d];
    }
#endif
    for (int i = tid; i < STEPS; i += 256) {
        usum[i] = usumg[(size_t)b * STEPS + i];
        msk[i]  = maskg[(size_t)b * STEPS + i];
    }
    for (int i = tid; i < 128 * 129; i += 256) blog[i] = 0.0f;
    __syncthreads();

    for (int r = 0; r < 3; ++r) {
        // ---- c = softmax over n (per s) * mask, in place in blog ----------
        if (tid < STEPS) {
            const int s = tid;
            float mx = -3.4e38f;
            for (int n = 0; n < NCAP; ++n) mx = fmaxf(mx, blog[n * 129 + s]);
            float sumv = 0.0f;
            for (int n = 0; n < NCAP; ++n) sumv += __expf(blog[n * 129 + s] - mx);
            const float inv = msk[s] / sumv;
            for (int n = 0; n < NCAP; ++n)
                blog[n * 129 + s] = __expf(blog[n * 129 + s] - mx) * inv;
        }
        __syncthreads();

        // ---- t2[n] = sum_s c[n,s] * usum[s] -------------------------------
        if (tid < NCAP) {
            const int n = tid;
            float acc = 0.0f;
            for (int s = 0; s < STEPS; ++s) acc += blog[n * 129 + s] * usum[s];
            t2[n] = acc;
        }

        // ---- t1 = C(128x128) @ UW(128x64) via WMMA ------------------------
        {
            const int m0 = wave * 16;
            v8f acc[4] = {};
            for (int k0 = 0; k0 < STEPS; k0 += 4) {
                const int ka = k0 + 2 * khi;
                v2f a;
                a.x = blog[(m0 + mrow) * 129 + ka];
                a.y = blog[(m0 + mrow) * 129 + ka + 1];
#pragma unroll
                for (int t = 0; t < 4; ++t) {
                    v2f bb;
                    bb.x = uw[ka * 65 + t * 16 + mrow];
                    bb.y = uw[(ka + 1) * 65 + t * 16 + mrow];
                    acc[t] = wmma_f32(a, bb, acc[t]);
                }
            }
#pragma unroll
            for (int t = 0; t < 4; ++t)
#pragma unroll
                for (int j = 0; j < 8; ++j)
                    pre[(m0 + j + 8 * khi) * 65 + t * 16 + mrow] = acc[t][j];
        }
        __syncthreads();

        // ---- pre += t2[n]*pe1 + sum_s c[n,s]*pe2[s, n*64+d] ---------------
        for (int i = tid; i < NCAP * DCAP; i += 256) {
            const int n = i >> 6, d = i & 63;
            const float* p2 = pe2g + (size_t)(n * DCAP + d);
            float acc = 0.0f;
            for (int s = 0; s < STEPS; ++s) {
                __builtin_prefetch(p2 + (size_t)(s + 8) * (NCAP * DCAP), 0, 0);
                acc += blog[n * 129 + s] * p2[(size_t)s * (NCAP * DCAP)];
            }
            pre[n * 65 + d] += t2[n] * pe1g[i] + acc;
        }
        __syncthreads();

        // ---- squash -> outm ----------------------------------------------
        if (tid < NCAP) {
            const int n = tid;
            float ss = 0.0f;
            for (int d = 0; d < DCAP; ++d) {
                float v = pre[n * 65 + d];
                ss += v * v;
            }
            const float scale = ss / (1.0f + ss) * __frsqrt_rn(ss + EPSF);
            for (int d = 0; d < DCAP; ++d)
                outm[n * 65 + d] = scale * pre[n * 65 + d];
        }
        __syncthreads();

        if (r < 2) {
            // ---- g2[n] = sum_d o[n,d]*pe1[n,d] ----------------------------
            if (tid < NCAP) {
                const int n = tid;
                float acc = 0.0f;
                for (int d = 0; d < DCAP; ++d)
                    acc += outm[n * 65 + d] * pe1g[n * DCAP + d];
                g2[n] = acc;
            }
            // ---- g1 = O(128x64) @ UW^T(64x128) via WMMA, overwrite blog ---
            {
                const int m0 = wave * 16;
                v8f acc[8] = {};
                for (int k0 = 0; k0 < DCAP; k0 += 4) {
                    const int ka = k0 + 2 * khi;
                    v2f a;
                    a.x = outm[(m0 + mrow) * 65 + ka];
                    a.y = outm[(m0 + mrow) * 65 + ka + 1];
#pragma unroll
                    for (int t = 0; t < 8; ++t) {
                        v2f bb;  // B[k=d][n=s] = uw[s][d]
                        bb.x = uw[(t * 16 + mrow) * 65 + ka];
                        bb.y = uw[(t * 16 + mrow) * 65 + ka + 1];
                        acc[t] = wmma_f32(a, bb, acc[t]);
                    }
                }
#pragma unroll
                for (int t = 0; t < 8; ++t)
#pragma unroll
                    for (int j = 0; j < 8; ++j)
                        blog[(m0 + j + 8 * khi) * 129 + t * 16 + mrow] = acc[t][j];
            }
            __syncthreads();
            // ---- blog += usum[s]*g2[n] + sum_d o[n,d]*pe2[s,n*64+d] -------
            for (int i = tid; i < NCAP * STEPS; i += 256) {
                const int n = i >> 7, s = i & 127;
                const float* p2 = pe2g + (size_t)s * (NCAP * DCAP) + n * DCAP;
                float acc = 0.0f;
                for (int d = 0; d < DCAP; ++d) acc += outm[n * 65 + d] * p2[d];
                blog[n * 129 + s] += usum[s] * g2[n] + acc;
            }
            __syncthreads();
        }
    }

    for (int i = tid; i < NCAP * DCAP; i += 256) {
        const int n = i >> 6, d = i & 63;
        out[(size_t)b * (NCAP * DCAP) + i] = outm[n * 65 + d];
    }
}

extern "C" void kernel_launch(void* const* d_in, const int* in_sizes, int n_in,
                              void* d_out, int out_size, void* d_ws, size_t ws_size,
                              hipStream_t stream) {
    const float* u    = (const float*)d_in[0];   // (32,128,256)
    const float* mask = (const float*)d_in[1];   // (32,128)
    const float* W    = (const float*)d_in[2];   // (1,256,64)
    float* out = (float*)d_out;                  // (32,128,64)

    float* ws    = (float*)d_ws;
    float* pe1   = ws;                                   // 8192
    float* pe2   = pe1 + NCAP * DCAP;                    // 1,048,576
    float* UWg   = pe2 + STEPS * NCAP * DCAP;            // 262,144
    float* usumg = UWg + (size_t)BATCH * STEPS * DCAP;   // 4,096

    pe1_kernel<<<(NCAP * DCAP + 255) / 256, 256, 0, stream>>>(pe1);
    pe2_kernel<<<(STEPS * NCAP * DCAP + 255) / 256, 256, 0, stream>>>(pe2);
    gemm_uw_kernel<<<BATCH * STEPS / 16, 32, 0, stream>>>(u, W, UWg, usumg);

    const size_t smem = (size_t)(128 * 65 * 3 + 128 * 129 + 4 * 128) * sizeof(float);
    routing_kernel<<<BATCH, 256, smem, stream>>>(UWg, usumg, pe1, pe2, mask, out);
}